// GCNRegression_3504693313558
// MI455X (gfx1250) — compile-verified
//
#include <hip/hip_runtime.h>

#define NN 200000
#define EE 3200000
#define GG 1024
#define HH 16
#define ALPHA 0.2f

typedef float v2f __attribute__((ext_vector_type(2)));
typedef float v8f __attribute__((ext_vector_type(8)));

// ---------------- elementwise / scatter kernels ----------------

__global__ void k_fill(float* p, float v, int n) {
    int i = blockIdx.x * blockDim.x + threadIdx.x;
    if (i < n) p[i] = v;
}

__global__ void k_scatter_deg(const int* __restrict__ dst, const float* __restrict__ ea,
                              float* __restrict__ deg, int e) {
    int i = blockIdx.x * blockDim.x + threadIdx.x;
    if (i < e) atomicAdd(&deg[dst[i]], ea[i]);
}

__global__ void k_dinv(float* __restrict__ deg, int n) {
    int i = blockIdx.x * blockDim.x + threadIdx.x;
    if (i < n) deg[i] = rsqrtf(deg[i]);
}

__global__ void k_enorm(const int* __restrict__ src, const int* __restrict__ dst,
                        const float* __restrict__ ea, const float* __restrict__ dinv,
                        float* __restrict__ enorm, int e) {
    int i = blockIdx.x * blockDim.x + threadIdx.x;
    if (i < e) enorm[i] = dinv[src[i]] * ea[i] * dinv[dst[i]];
}

// layer 0: hW = x (N,1) * cw0 (1,16); also zero agg
__global__ void k_layer0(const float* __restrict__ x, const float* __restrict__ cw0,
                         float* __restrict__ hW, float* __restrict__ agg, int n16) {
    int t = blockIdx.x * blockDim.x + threadIdx.x;
    if (t < n16) {
        int j = t & 15;
        int node = t >> 4;
        hW[t] = x[node] * cw0[j];
        agg[t] = 0.0f;
    }
}

// 16 threads per edge: half-wave gathers one coalesced hW row, scatter-adds to agg[dst]
__global__ void k_scatter_edges(const int* __restrict__ src, const int* __restrict__ dst,
                                const float* __restrict__ enorm, const float* __restrict__ hW,
                                float* __restrict__ agg, int e) {
    int t = blockIdx.x * blockDim.x + threadIdx.x;
    int ed = t >> 4;
    if (ed < e) {
        int j = t & 15;
        float v = enorm[ed] * hW[src[ed] * 16 + j];
        atomicAdd(&agg[dst[ed] * 16 + j], v);
    }
}

// agg := relu(agg + dinv^2 * hW + cb)   (t matrix, in place)
__global__ void k_combine(const float* __restrict__ hW, const float* __restrict__ dinv,
                          const float* __restrict__ cb_i, float* __restrict__ agg, int n16) {
    int t = blockIdx.x * blockDim.x + threadIdx.x;
    if (t < n16) {
        int j = t & 15;
        int node = t >> 4;
        float d = dinv[node];
        float v = agg[t] + d * d * hW[t] + cb_i[j];
        agg[t] = fmaxf(v, 0.0f);
    }
}

// ---------------- WMMA fp32 GEMM: out[rows,16] = in[rows,16] @ W[16,16] (+bias)(+relu) ----
// One wave per 16-row tile; K=16 split into 4 chained V_WMMA_F32_16X16X4_F32.
// Optionally zero-fills zero_buf over the same extent (free sideband pass).
__global__ void k_gemm16(const float* __restrict__ in, const float* __restrict__ W,
                         const float* __restrict__ bias, float* __restrict__ out,
                         float* __restrict__ zero_buf, int tiles, int relu) {
    int lane = threadIdx.x & 31;
    int wave_in_block = threadIdx.x >> 5;
    int waves_per_block = blockDim.x >> 5;
    int wave = blockIdx.x * waves_per_block + wave_in_block;
    int col = lane & 15;
    int half = lane >> 4;   // 0: lanes 0-15, 1: lanes 16-31

    // B (4x16 f32) per K-chunk: VGPR v holds K=v (lanes 0-15) / K=v+2 (lanes 16-31)
    v2f b[4];
#pragma unroll
    for (int k = 0; k < 4; ++k) {
        int kr = k * 4 + half * 2;
        b[k].x = W[(kr + 0) * 16 + col];
        b[k].y = W[(kr + 1) * 16 + col];
    }
    float bias_c = bias ? bias[col] : 0.0f;

    for (int tile = wave; tile < tiles; tile += gridDim.x * waves_per_block) {
        const float* rowp = in + (size_t)tile * 256;
        v8f c;
#pragma unroll
        for (int r = 0; r < 8; ++r) c[r] = bias_c;  // bias depends on N only -> broadcast rows
#pragma unroll
        for (int k = 0; k < 4; ++k) {
            // A (16x4 f32): lanes 0-15 row=lane K={0,1}; lanes 16-31 row=lane-16 K={2,3}
            int kc = k * 4 + half * 2;
            v2f a;
            a.x = rowp[col * 16 + kc + 0];
            a.y = rowp[col * 16 + kc + 1];
            c = __builtin_amdgcn_wmma_f32_16x16x4_f32(false, a, false, b[k],
                                                      (short)0, c, false, false);
        }
        float* op = out + (size_t)tile * 256;
#pragma unroll
        for (int r = 0; r < 8; ++r) {
            float v = c[r];
            if (relu) v = fmaxf(v, 0.0f);
            op[(r + half * 8) * 16 + col] = v;   // D: M = r + 8*half, N = col
        }
        if (zero_buf) {
            float* zp = zero_buf + (size_t)tile * 256;
#pragma unroll
            for (int r = 0; r < 8; ++r) zp[(r + half * 8) * 16 + col] = 0.0f;
        }
    }
}

// ---------------- attention pooling ----------------

__global__ void k_pool_acc(const float* __restrict__ h, const int* __restrict__ batch,
                           float* __restrict__ denom, float* __restrict__ num, int n16) {
    int t = blockIdx.x * blockDim.x + threadIdx.x;
    if (t < n16) {
        int j = t & 15;
        int node = t >> 4;
        int g = batch[node];
        float hv = h[t];
        float s = expf(ALPHA * hv);
        atomicAdd(&denom[g * 16 + j], s);
        atomicAdd(&num[g * 16 + j], hv * s);
    }
}

__global__ void k_pooled(const float* __restrict__ num, const float* __restrict__ denom,
                         float* __restrict__ pooled, int n) {
    int i = blockIdx.x * blockDim.x + threadIdx.x;
    if (i < n) pooled[i] = num[i] / denom[i];
}

// out[g] = m2[g,:] . w3 + b3
__global__ void k_final(const float* __restrict__ m2, const float* __restrict__ w3,
                        const float* __restrict__ b3, float* __restrict__ out, int g) {
    int i = blockIdx.x * blockDim.x + threadIdx.x;
    if (i < g) {
        float acc = b3[0];
#pragma unroll
        for (int k = 0; k < 16; ++k) acc += m2[i * 16 + k] * w3[k];
        out[i] = acc;
    }
}

// ---------------- host launcher ----------------

extern "C" void kernel_launch(void* const* d_in, const int* in_sizes, int n_in,
                              void* d_out, int out_size, void* d_ws, size_t ws_size,
                              hipStream_t stream) {
    const float* x   = (const float*)d_in[0];
    const int*   ei  = (const int*)d_in[1];
    const int*   bat = (const int*)d_in[2];
    const float* ea  = (const float*)d_in[3];
    const float* cw0 = (const float*)d_in[4];
    const float* cwr = (const float*)d_in[5];   // [L-1,16,16]
    const float* cb  = (const float*)d_in[6];   // [L,16]
    const float* lw  = (const float*)d_in[7];   // [L,16,16]
    const float* lb  = (const float*)d_in[8];   // [L,16]
    const float* w1  = (const float*)d_in[9];
    const float* b1  = (const float*)d_in[10];
    const float* w2  = (const float*)d_in[11];
    const float* b2  = (const float*)d_in[12];
    const float* w3  = (const float*)d_in[13];
    const float* b3  = (const float*)d_in[14];
    float* out = (float*)d_out;

    const int* src = ei;
    const int* dst = ei + EE;

    // workspace layout (floats); all region sizes are multiples of 64
    float* ws     = (float*)d_ws;
    float* dinv   = ws;                       // N
    float* enorm  = dinv + NN;                // E
    float* hW     = enorm + EE;               // N*16
    float* agg    = hW + (size_t)NN * 16;     // N*16 (also holds t in place)
    float* h      = agg + (size_t)NN * 16;    // N*16
    float* denom  = h + (size_t)NN * 16;      // G*16
    float* num    = denom + GG * 16;          // G*16 (contiguous with denom)
    float* pooled = num + GG * 16;            // G*16
    float* m1     = pooled + GG * 16;         // G*16
    float* m2     = m1 + GG * 16;             // G*16

    const int B = 256;
    const int gN   = (NN + B - 1) / B;
    const int gE   = (EE + B - 1) / B;
    const int gN16 = (NN * 16 + B - 1) / B;
    const int gE16 = ((EE * 16) + B - 1) / B;
    const int tilesN = NN / 16;               // 12500 exactly
    const int tilesG = GG / 16;               // 64
    const int wpb = B / 32;                   // 8 waves per block
    const int gGemmN = (tilesN + wpb - 1) / wpb;
    const int gGemmG = (tilesG + wpb - 1) / wpb;

    // degree -> dinv
    k_fill<<<gN, B, 0, stream>>>(dinv, 1.0f, NN);
    k_scatter_deg<<<gE, B, 0, stream>>>(dst, ea, dinv, EE);
    k_dinv<<<gN, B, 0, stream>>>(dinv, NN);
    k_enorm<<<gE, B, 0, stream>>>(src, dst, ea, dinv, enorm, EE);

    // ---- layer 0 ----
    k_layer0<<<gN16, B, 0, stream>>>(x, cw0, hW, agg, NN * 16);
    k_scatter_edges<<<gE16, B, 0, stream>>>(src, dst, enorm, hW, agg, EE);
    k_combine<<<gN16, B, 0, stream>>>(hW, dinv, cb + 0 * 16, agg, NN * 16);
    k_gemm16<<<gGemmN, B, 0, stream>>>(agg, lw + 0 * 256, lb + 0 * 16, h, nullptr, tilesN, 0);

    // ---- layers 1..2 ----
    for (int i = 1; i < 3; ++i) {
        // hW = h @ cwr[i-1]; zero agg as sideband
        k_gemm16<<<gGemmN, B, 0, stream>>>(h, cwr + (size_t)(i - 1) * 256, nullptr,
                                           hW, agg, tilesN, 0);
        k_scatter_edges<<<gE16, B, 0, stream>>>(src, dst, enorm, hW, agg, EE);
        k_combine<<<gN16, B, 0, stream>>>(hW, dinv, cb + i * 16, agg, NN * 16);
        k_gemm16<<<gGemmN, B, 0, stream>>>(agg, lw + (size_t)i * 256, lb + i * 16,
                                           h, nullptr, tilesN, 0);
    }

    // ---- attention pooling ----
    k_fill<<<(2 * GG * 16 + B - 1) / B, B, 0, stream>>>(denom, 0.0f, 2 * GG * 16);
    k_pool_acc<<<gN16, B, 0, stream>>>(h, bat, denom, num, NN * 16);
    k_pooled<<<(GG * 16 + B - 1) / B, B, 0, stream>>>(num, denom, pooled, GG * 16);

    // ---- final MLP ----
    k_gemm16<<<gGemmG, B, 0, stream>>>(pooled, w1, b1, m1, nullptr, tilesG, 1);
    k_gemm16<<<gGemmG, B, 0, stream>>>(m1, w2, b2, m2, nullptr, tilesG, 1);
    k_final<<<(GG + B - 1) / B, B, 0, stream>>>(m2, w3, b3, out, GG);
}